// VectorEstimationLoss_25563645345928
// MI455X (gfx1250) — compile-verified
//
#include <hip/hip_runtime.h>
#include <math.h>
#include <float.h>

// ---------------------------------------------------------------------------
// VectorEstimationLoss on MI455X (gfx1250, wave32)
//   Stage 1: FPS   — serial chain on one workgroup; min_d register-resident,
//                    argmax via wave32 shuffle reduce (2 barriers/round)
//   Stage 2: KNN   — V_WMMA_F32_16X16X4_F32 cross-terms, fused top-20
//                    selection; d2 matrix never materialized, pos L2-resident
//   Stage 3: 3x3 covariance top-eigenvector (power iter) + log-abs-cos loss
// ---------------------------------------------------------------------------

#define N_PTS       40000
#define N_ANCH      4000          // ceil(0.1 * 40000)
#define KNN_K       20
#define FPS_THREADS 1024
#define FPS_STRIDE  1024
#define FPS_MAIN    39            // 1024*39 = 39936 always-valid stripe elems
#define FPS_TAILTHR 64            // threads 0..63 own one tail element (j=39)
#define COS_EPS     1e-8f
#define LOG_EPS     1e-6f

typedef __attribute__((ext_vector_type(2))) float v2f;
typedef __attribute__((ext_vector_type(8))) float v8f;

// ---- workspace layout (bytes) ----
#define WS_ANCHOR_POS   0u          // float4[N_ANCH]   64000 B (x,y,z,|p|^2)
#define WS_NNIDX        65536u      // int[N_ANCH*KNN_K] 320000 B
// total ~386 KB

// ---------------------------------------------------------------------------
__global__ __launch_bounds__(1) void init_out_kernel(float* out) {
    out[0] = 0.0f;
}

// ---------------------------------------------------------------------------
// Farthest point sampling. min_d lives in a fully-unrolled 40-register stripe
// per thread (no memory traffic for the scan/relax phases). Argmax per round:
// per-thread register scan -> wave32 shuffle reduce -> 32 wave leaders ->
// wave-0 shuffle reduce. Two __syncthreads per round total.
// Tie-break: lowest global index on equal value (matches jnp.argmax).
// ---------------------------------------------------------------------------
__global__ __launch_bounds__(FPS_THREADS)
void fps_kernel(const float* __restrict__ pos,
                float4* __restrict__ anchor_pos) {
    __shared__ float swv[FPS_THREADS / 32];
    __shared__ int   swi[FPS_THREADS / 32];
    __shared__ float scx, scy, scz;

    const int tid  = threadIdx.x;
    const int wave = tid >> 5;
    const int lane = tid & 31;
    const bool has_tail = (tid < FPS_TAILTHR);

    if (tid == 0) {
        float x = pos[0], y = pos[1], z = pos[2];
        scx = x; scy = y; scz = z;
        anchor_pos[0] = make_float4(x, y, z, x * x + y * y + z * z);
    }
    __syncthreads();

    // ---- init register-resident min_d stripe: dist^2 to point 0 ----
    float md[FPS_MAIN + 1];
    {
        float px = scx, py = scy, pz = scz;
#pragma unroll
        for (int j = 0; j < FPS_MAIN; ++j) {
            int i = tid + j * FPS_STRIDE;
            float dx = pos[3 * i] - px, dy = pos[3 * i + 1] - py, dz = pos[3 * i + 2] - pz;
            md[j] = dx * dx + dy * dy + dz * dz;
        }
        md[FPS_MAIN] = -FLT_MAX;           // tail slot: never selected if invalid
        if (has_tail) {
            int i = tid + FPS_MAIN * FPS_STRIDE;
            float dx = pos[3 * i] - px, dy = pos[3 * i + 1] - py, dz = pos[3 * i + 2] - pz;
            md[FPS_MAIN] = dx * dx + dy * dy + dz * dz;
        }
    }
    __syncthreads();

    for (int it = 1; it < N_ANCH; ++it) {
        // ---- per-thread register argmax ('>' keeps first => lowest index) --
        float best = md[0]; int bj = 0;
#pragma unroll
        for (int j = 1; j <= FPS_MAIN; ++j)
            if (md[j] > best) { best = md[j]; bj = j; }
        int bidx = tid + bj * FPS_STRIDE;

        // ---- wave32 shuffle argmax reduce (ds permute, no barrier) ----
#pragma unroll
        for (int off = 16; off > 0; off >>= 1) {
            float ov = __shfl_xor(best, off, 32);
            int   oi = __shfl_xor(bidx, off, 32);
            if (ov > best || (ov == best && oi < bidx)) { best = ov; bidx = oi; }
        }
        if (lane == 0) { swv[wave] = best; swi[wave] = bidx; }
        __syncthreads();

        // ---- cross-wave reduce by wave 0 (32 partials == 32 lanes) ----
        if (wave == 0) {
            float bv = swv[lane]; int bi = swi[lane];
#pragma unroll
            for (int off = 16; off > 0; off >>= 1) {
                float ov = __shfl_xor(bv, off, 32);
                int   oi = __shfl_xor(bi, off, 32);
                if (ov > bv || (ov == bv && oi < bi)) { bv = ov; bi = oi; }
            }
            if (lane == 0) {
                float x = pos[3 * bi], y = pos[3 * bi + 1], z = pos[3 * bi + 2];
                scx = x; scy = y; scz = z;
                anchor_pos[it] = make_float4(x, y, z, x * x + y * y + z * z);
            }
        }
        __syncthreads();

        // ---- relax stripe against new anchor (pure VALU + L0-hot loads) ----
        float px = scx, py = scy, pz = scz;
#pragma unroll
        for (int j = 0; j < FPS_MAIN; ++j) {
            int i = tid + j * FPS_STRIDE;
            float dx = pos[3 * i] - px, dy = pos[3 * i + 1] - py, dz = pos[3 * i + 2] - pz;
            float d = dx * dx + dy * dy + dz * dz;
            md[j] = fminf(md[j], d);
        }
        if (has_tail) {
            int i = tid + FPS_MAIN * FPS_STRIDE;
            float dx = pos[3 * i] - px, dy = pos[3 * i + 1] - py, dz = pos[3 * i + 2] - pz;
            float d = dx * dx + dy * dy + dz * dz;
            md[FPS_MAIN] = fminf(md[FPS_MAIN], d);
        }
        // no barrier needed here: stripe is thread-private
    }
}

// ---------------------------------------------------------------------------
// Fused KNN. One wave (32 lanes) per 32 anchors; 2500 point tiles of 16.
// Cross term a.p via V_WMMA_F32_16X16X4_F32 (K=4: xyz + 0 pad). Selection
// key = |p|^2 - 2 a.p (per-row |a|^2 constant dropped — cannot change the
// top-K ordering). |p|^2 computed in-lane (3 FLOPs beat a global load).
// WMMA C-tile transposed through a per-wave LDS buffer so each lane owns one
// anchor's register-resident sorted top-20 (unrolled bubble insert).
// ---------------------------------------------------------------------------
__global__ __launch_bounds__(32)
void knn_kernel(const float* __restrict__ pos,
                const float4* __restrict__ anchor_pos,
                int* __restrict__ nn_idx) {
    __shared__ float keys[32][16];   // [anchor within group][point within tile]

    const int lane = threadIdx.x;          // one full wave, EXEC all 1s
    const int half = lane >> 4;            // 0: K=0,1   1: K=2,3
    const int l    = lane & 15;
    const int g    = blockIdx.x * 32;      // 4000/32 = 125 blocks

    float4 a0 = anchor_pos[g + l];
    float4 a1 = anchor_pos[g + 16 + l];
    v2f A0, A1;
    A0.x = half ? a0.z : a0.x;  A0.y = half ? 0.0f : a0.y;
    A1.x = half ? a1.z : a1.x;  A1.y = half ? 0.0f : a1.y;

    float kd[KNN_K];
    int   ki[KNN_K];
#pragma unroll
    for (int j = 0; j < KNN_K; ++j) { kd[j] = FLT_MAX; ki[j] = -1; }

    for (int pt = 0; pt < N_PTS; pt += 16) {
        const int pi = pt + l;
        float px = pos[3 * pi], py = pos[3 * pi + 1], pz = pos[3 * pi + 2];
        v2f B;
        B.x = half ? pz : px;  B.y = half ? 0.0f : py;
        float pn = px * px + py * py + pz * pz;   // |p|^2 for this column

        if (pt + 256 < N_PTS)   // stream-ahead: global_prefetch_b8
            __builtin_prefetch(&pos[3 * (pt + 256)], 0, 1);

        v8f c0 = {};
        v8f c1 = {};
        c0 = __builtin_amdgcn_wmma_f32_16x16x4_f32(false, A0, false, B,
                                                   (short)0, c0, false, false);
        c1 = __builtin_amdgcn_wmma_f32_16x16x4_f32(false, A1, false, B,
                                                   (short)0, c1, false, false);

        const int mrow = 8 * half;
#pragma unroll
        for (int v = 0; v < 8; ++v) {
            keys[mrow + v][l]      = pn - 2.0f * c0[v];
            keys[16 + mrow + v][l] = pn - 2.0f * c1[v];
        }
        __syncthreads();

#pragma unroll 4
        for (int j = 0; j < 16; ++j) {
            float d = keys[lane][j];
            if (d < kd[KNN_K - 1]) {
                float cd = d; int ci = pt + j;
#pragma unroll
                for (int s = 0; s < KNN_K; ++s) {
                    bool sw = cd < kd[s];
                    float td = sw ? kd[s] : cd;  int ti = sw ? ki[s] : ci;
                    kd[s] = sw ? cd : kd[s];     ki[s] = sw ? ci : ki[s];
                    cd = td; ci = ti;
                }
            }
        }
        __syncthreads();
    }

#pragma unroll
    for (int j = 0; j < KNN_K; ++j)
        nn_idx[(g + lane) * KNN_K + j] = ki[j];
}

// ---------------------------------------------------------------------------
// Per anchor: 3x3 covariance of its 20 neighbors, dominant eigenvector by
// power iteration (== top right singular vector vh[:,0,:]; sign irrelevant
// under abs-cos), then -log(|cos|+eps)/N_ANCH. Reference's idx0 quirk indexes
// vec_pred by anchor ORDINAL; the K-fold repeat is a no-op under the mean.
// ---------------------------------------------------------------------------
__global__ __launch_bounds__(256)
void eigen_loss_kernel(const float* __restrict__ pos,
                       const float* __restrict__ vec_pred,
                       const int* __restrict__ nn_idx,
                       float* __restrict__ out) {
    __shared__ float red[256];
    const int a = blockIdx.x * blockDim.x + threadIdx.x;
    float contrib = 0.0f;

    if (a < N_ANCH) {
        float sx = 0, sy = 0, sz = 0;
        float sxx = 0, sxy = 0, sxz = 0, syy = 0, syz = 0, szz = 0;
#pragma unroll 4
        for (int j = 0; j < KNN_K; ++j) {
            int i = nn_idx[a * KNN_K + j];
            float x = pos[3 * i], y = pos[3 * i + 1], z = pos[3 * i + 2];
            sx += x; sy += y; sz += z;
            sxx += x * x; sxy += x * y; sxz += x * z;
            syy += y * y; syz += y * z; szz += z * z;
        }
        const float inv_k = 1.0f / (float)KNN_K;
        float cxx = sxx - sx * sx * inv_k;
        float cxy = sxy - sx * sy * inv_k;
        float cxz = sxz - sx * sz * inv_k;
        float cyy = syy - sy * sy * inv_k;
        float cyz = syz - sy * sz * inv_k;
        float czz = szz - sz * sz * inv_k;

        float vx = 1.0f, vy = 1.0f, vz = 1.0f;
#pragma unroll 8
        for (int it = 0; it < 48; ++it) {
            float nx = cxx * vx + cxy * vy + cxz * vz;
            float ny = cxy * vx + cyy * vy + cyz * vz;
            float nz = cxz * vx + cyz * vy + czz * vz;
            float rn = rsqrtf(nx * nx + ny * ny + nz * nz + 1e-30f);
            vx = nx * rn; vy = ny * rn; vz = nz * rn;
        }

        float ax = vec_pred[3 * a], ay = vec_pred[3 * a + 1], az = vec_pred[3 * a + 2];
        float na = fmaxf(sqrtf(ax * ax + ay * ay + az * az), COS_EPS);
        float nb = fmaxf(sqrtf(vx * vx + vy * vy + vz * vz), COS_EPS);
        float cosv = fabsf(ax * vx + ay * vy + az * vz) / (na * nb);
        contrib = -logf(cosv + LOG_EPS) * (1.0f / (float)N_ANCH);
    }

    red[threadIdx.x] = contrib;
    __syncthreads();
    for (int s = 128; s > 0; s >>= 1) {
        if (threadIdx.x < s) red[threadIdx.x] += red[threadIdx.x + s];
        __syncthreads();
    }
    if (threadIdx.x == 0) atomicAdd(out, red[0]);
}

// ---------------------------------------------------------------------------
extern "C" void kernel_launch(void* const* d_in, const int* in_sizes, int n_in,
                              void* d_out, int out_size, void* d_ws, size_t ws_size,
                              hipStream_t stream) {
    const float* vec_pred = (const float*)d_in[0];   // [40000,3] f32
    const float* pos      = (const float*)d_in[1];   // [40000,3] f32
    float* out = (float*)d_out;                      // scalar f32

    char* ws = (char*)d_ws;
    float4* anch_pos = (float4*)(ws + WS_ANCHOR_POS);
    int*    nn       = (int*)   (ws + WS_NNIDX);

    init_out_kernel<<<1, 1, 0, stream>>>(out);
    fps_kernel<<<1, FPS_THREADS, 0, stream>>>(pos, anch_pos);
    knn_kernel<<<N_ANCH / 32, 32, 0, stream>>>(pos, anch_pos, nn);
    eigen_loss_kernel<<<(N_ANCH + 255) / 256, 256, 0, stream>>>(pos, vec_pred, nn, out);
}